// GCN_7060926234938
// MI455X (gfx1250) — compile-verified
//
#include <hip/hip_runtime.h>
#include <hip/hip_bf16.h>
#include <math.h>

// ---------------- problem constants ----------------
#define N_PROC 8000
#define N_PAT  8000
#define N_DOC  4000
#define NN     20000          // total nodes
#define RR     2048           // reg_num
#define EE     640000         // edges (before self loops)
#define ETOT   (EE + NN)      // edges + self loops
#define NEG_SLOPE 0.2f

typedef float v2f __attribute__((ext_vector_type(2)));
typedef float v8f __attribute__((ext_vector_type(8)));

// ---------------- helpers ----------------
__device__ __forceinline__ unsigned fenc(float f) {
    unsigned u = __float_as_uint(f);
    return (u & 0x80000000u) ? ~u : (u | 0x80000000u);
}
__device__ __forceinline__ float fdec(unsigned u) {
    return __uint_as_float((u & 0x80000000u) ? (u & 0x7fffffffu) : ~u);
}
__device__ __forceinline__ unsigned hashrng(unsigned a, unsigned b) {
    unsigned x = a * 0x9E3779B1u + b * 0x85EBCA77u + 0x165667B1u;
    x ^= x >> 15; x *= 0x2C1B3C6Du;
    x ^= x >> 12; x *= 0x297A2D39u;
    x ^= x >> 15;
    return x;
}

// ---------------- embedding: x0[N,4] (stride 8, zero padded) ----------------
__global__ void k_embed(const float* __restrict__ proc, const float* __restrict__ pat,
                        const float* __restrict__ doc,
                        const float* __restrict__ W_ep, const float* __restrict__ b_ep,
                        const float* __restrict__ W_pp, const float* __restrict__ b_pp,
                        const float* __restrict__ W_dp, const float* __restrict__ b_dp,
                        float* __restrict__ X) {
    int n = blockIdx.x * blockDim.x + threadIdx.x;
    if (n >= NN) return;
    float o[4];
    if (n < N_PROC) {
        const float* in = proc + n * 4;
        for (int j = 0; j < 4; ++j) {
            float a = b_ep[j];
            for (int i = 0; i < 4; ++i) a += in[i] * W_ep[i * 4 + j];
            o[j] = a;
        }
    } else if (n < N_PROC + N_PAT) {
        int r = n - N_PROC;
        float v = pat[r];
        for (int j = 0; j < 4; ++j) o[j] = b_pp[j] + v * W_pp[j];
    } else {
        int r = n - N_PROC - N_PAT;
        const float* in = doc + r * 2;
        for (int j = 0; j < 4; ++j) {
            float a = b_dp[j];
            for (int i = 0; i < 2; ++i) a += in[i] * W_dp[i * 4 + j];
            o[j] = a;
        }
    }
    float* xo = X + (size_t)n * 8;
    for (int j = 0; j < 4; ++j) xo[j] = o[j];
    for (int j = 4; j < 8; ++j) xo[j] = 0.f;
}

// ---------------- GAT per-node: xh = x@W, attention logits, init accumulators ----
__global__ void k_node_prep(const float* __restrict__ X, const float* __restrict__ W,
                            const float* __restrict__ a_s, const float* __restrict__ a_d,
                            float* __restrict__ XH, float* __restrict__ ALS,
                            float* __restrict__ ALD, unsigned* __restrict__ M,
                            float* __restrict__ S, float* __restrict__ ACC,
                            int in_dim, int H, int C) {
    int n = blockIdx.x * blockDim.x + threadIdx.x;
    if (n >= NN) return;
    float xi[8], hv[8];
    const float* xr = X + (size_t)n * 8;
    for (int i = 0; i < in_dim; ++i) xi[i] = xr[i];
    int HC = H * C;
    for (int j = 0; j < HC; ++j) {
        float acc = 0.f;
        for (int i = 0; i < in_dim; ++i) acc += xi[i] * W[i * HC + j];
        hv[j] = acc;
        XH[(size_t)n * 8 + j] = acc;
    }
    for (int h = 0; h < H; ++h) {
        float s = 0.f, d = 0.f;
        for (int c = 0; c < C; ++c) {
            float v = hv[h * C + c];
            s += v * a_s[h * C + c];
            d += v * a_d[h * C + c];
        }
        ALS[(size_t)n * 4 + h] = s;
        ALD[(size_t)n * 4 + h] = d;
        M[(size_t)n * 4 + h] = 0u;   // encoded minimum
        S[(size_t)n * 4 + h] = 0.f;
    }
    for (int j = 0; j < 8; ++j) ACC[(size_t)n * 8 + j] = 0.f;
}

// ---------------- edge pass 1: segment max of attention logits -----------------
__global__ void k_edge_max(const int* __restrict__ ei, const float* __restrict__ ALS,
                           const float* __restrict__ ALD, unsigned* __restrict__ M, int H) {
    int e = blockIdx.x * blockDim.x + threadIdx.x;
    if (e >= ETOT) return;
    int s = (e < EE) ? ei[e] : (e - EE);
    int d = (e < EE) ? ei[EE + e] : (e - EE);
    for (int h = 0; h < H; ++h) {
        float v = ALS[(size_t)s * 4 + h] + ALD[(size_t)d * 4 + h];
        v = (v > 0.f) ? v : NEG_SLOPE * v;
        atomicMax(&M[(size_t)d * 4 + h], fenc(v));
    }
}

// ---------------- edge pass 2: exp-sum + unnormalized weighted scatter ---------
__global__ void k_edge_acc(const int* __restrict__ ei, const float* __restrict__ ALS,
                           const float* __restrict__ ALD, const unsigned* __restrict__ M,
                           const float* __restrict__ XH, float* __restrict__ S,
                           float* __restrict__ ACC, int H, int C) {
    int e = blockIdx.x * blockDim.x + threadIdx.x;
    if (e >= ETOT) return;
    int s = (e < EE) ? ei[e] : (e - EE);
    int d = (e < EE) ? ei[EE + e] : (e - EE);
    for (int h = 0; h < H; ++h) {
        float v = ALS[(size_t)s * 4 + h] + ALD[(size_t)d * 4 + h];
        v = (v > 0.f) ? v : NEG_SLOPE * v;
        float mx = fdec(M[(size_t)d * 4 + h]);
        float a = __expf(v - mx);
        atomicAdd(&S[(size_t)d * 4 + h], a);
        for (int c = 0; c < C; ++c)
            atomicAdd(&ACC[(size_t)d * 8 + h * C + c], a * XH[(size_t)s * 8 + h * C + c]);
    }
}

// ---------------- finalize: normalize, bias, elu+dropout / dense out -----------
__global__ void k_finalize(const float* __restrict__ ACC, const float* __restrict__ S,
                           const float* __restrict__ b, float* __restrict__ XOUT,
                           int H, int C, int act_drop, unsigned seed, int dense) {
    int n = blockIdx.x * blockDim.x + threadIdx.x;
    if (n >= NN) return;
    int HC = H * C;
    for (int j = 0; j < HC; ++j) {
        float v = ACC[(size_t)n * 8 + j] / (S[(size_t)n * 4 + j / C] + 1e-16f) + b[j];
        if (act_drop) {
            v = (v > 0.f) ? v : expm1f(v);                 // ELU
            unsigned r = hashrng(seed, (unsigned)(n * HC + j));
            v = (r & 0x80000000u) ? 0.f : v * 2.0f;        // dropout p=0.5, scale 2
        }
        if (dense) XOUT[n] = v;                            // layer 3: HC==1
        else       XOUT[(size_t)n * 8 + j] = v;
    }
    if (!dense) for (int j = HC; j < 8; ++j) XOUT[(size_t)n * 8 + j] = 0.f;
}

// ---------------- GEMV out[j] = sum_n x[n]*W[n,j] via V_WMMA_F32_16X16X4_F32 ----
// 128 threads = 4 waves / block; each wave owns 16 columns, blockIdx.y slices rows.
// A operand: x[k..k+3] broadcast into every row of the 16x4 A tile, so all rows of
// D are equal and C[0] lanes 0..15 carry the 16 column dot-products.
__global__ void __launch_bounds__(128)
k_gemv_wmma(const float* __restrict__ x, const float* __restrict__ W,
            float* __restrict__ part, int ncols, int rows_per_slice) {
    __shared__ float lx[1024];
    int tid  = threadIdx.x;
    int wave = tid >> 5;
    int lane = tid & 31;
    int row0 = blockIdx.y * rows_per_slice;
    for (int i = tid; i < rows_per_slice; i += 128) lx[i] = x[row0 + i];
    __syncthreads();

    int col0  = (blockIdx.x * 4 + wave) * 16;
    int half2 = (lane >> 4) << 1;        // lanes 16..31 handle K+2
    int nc    = lane & 15;
    const float* wbase = W + (size_t)row0 * ncols + col0 + nc;

    v8f c = {0.f, 0.f, 0.f, 0.f, 0.f, 0.f, 0.f, 0.f};
    for (int k = 0; k < rows_per_slice; k += 4) {
        v2f a;
        a.x = lx[k + half2];             // A vgpr0: K = 0 / 2
        a.y = lx[k + half2 + 1];         // A vgpr1: K = 1 / 3
        const float* wp = wbase + (size_t)(k + half2) * ncols;
        v2f bm;
        bm.x = wp[0];                    // B vgpr0: K = 0 / 2, N = lane&15
        bm.y = wp[ncols];                // B vgpr1: K = 1 / 3
        c = __builtin_amdgcn_wmma_f32_16x16x4_f32(false, a, false, bm,
                                                  (short)0, c, false, false);
    }
    if (lane < 16) part[(size_t)blockIdx.y * ncols + col0 + lane] = c[0];
}

// ---------------- reduce partial slices + bias (+tanh) -------------------------
__global__ void k_reduce(const float* __restrict__ part, const float* __restrict__ bias,
                         float* __restrict__ out, int ncols, int nsl, int do_tanh) {
    int j = blockIdx.x * blockDim.x + threadIdx.x;
    if (j >= ncols) return;
    float s = bias[j];
    for (int i = 0; i < nsl; ++i) s += part[(size_t)i * ncols + j];
    out[j] = do_tanh ? tanhf(s) : s;
}

// ---------------- critic final dot: value = c1 . W_c2 + b_c2 -------------------
__global__ void __launch_bounds__(256)
k_value(const float* __restrict__ c1, const float* __restrict__ w,
        const float* __restrict__ b, float* __restrict__ out) {
    __shared__ float red[256];
    int tid = threadIdx.x;
    float s = 0.f;
    for (int j = tid; j < RR; j += 256) s += c1[j] * w[j];
    red[tid] = s;
    __syncthreads();
    for (int st = 128; st > 0; st >>= 1) {
        if (tid < st) red[tid] += red[tid + st];
        __syncthreads();
    }
    if (tid == 0) out[0] = red[0] + b[0];
}

// ---------------- masked softmax + Gumbel-max categorical sample ---------------
__global__ void __launch_bounds__(256)
k_policy(const float* __restrict__ h, const unsigned char* __restrict__ mask,
         float* __restrict__ probs, float* __restrict__ action_out) {
    __shared__ float rf[256];
    __shared__ float gv[256];
    __shared__ int   gi[256];
    int tid = threadIdx.x;
    const int PER = RR / 256;   // 8

    float p[PER];
    float lmax = -3.4e38f;
    for (int t = 0; t < PER; ++t) {
        int j = tid + t * 256;
        float v = mask[j] ? h[j] : -999999.0f;
        p[t] = v;
        lmax = fmaxf(lmax, v);
    }
    rf[tid] = lmax; __syncthreads();
    for (int st = 128; st > 0; st >>= 1) { if (tid < st) rf[tid] = fmaxf(rf[tid], rf[tid + st]); __syncthreads(); }
    float gmax = rf[0]; __syncthreads();

    float lsum = 0.f;
    for (int t = 0; t < PER; ++t) lsum += __expf(p[t] - gmax);
    rf[tid] = lsum; __syncthreads();
    for (int st = 128; st > 0; st >>= 1) { if (tid < st) rf[tid] += rf[tid + st]; __syncthreads(); }
    float gsum = rf[0]; __syncthreads();

    float bestv = -3.4e38f; int bestj = 0;
    for (int t = 0; t < PER; ++t) {
        int j = tid + t * 256;
        probs[j] = __expf(p[t] - gmax) / gsum;
        // Gumbel-max sampling from logits p
        unsigned r = hashrng(0x33330000u, (unsigned)j);
        float u = ((float)r + 1.0f) * 2.3283064e-10f;   // (0,1]
        float g = p[t] - logf(-logf(u));
        if (g > bestv) { bestv = g; bestj = j; }
    }
    gv[tid] = bestv; gi[tid] = bestj; __syncthreads();
    for (int st = 128; st > 0; st >>= 1) {
        if (tid < st) {
            if (gv[tid + st] > gv[tid] ||
                (gv[tid + st] == gv[tid] && gi[tid + st] < gi[tid])) {
                gv[tid] = gv[tid + st]; gi[tid] = gi[tid + st];
            }
        }
        __syncthreads();
    }
    if (tid == 0) action_out[0] = (float)gi[0];
}

// ---------------- host orchestration ------------------------------------------
extern "C" void kernel_launch(void* const* d_in, const int* in_sizes, int n_in,
                              void* d_out, int out_size, void* d_ws, size_t ws_size,
                              hipStream_t stream) {
    (void)in_sizes; (void)n_in; (void)out_size; (void)ws_size;
    const float* proc  = (const float*)d_in[0];
    const float* pat   = (const float*)d_in[1];
    const float* doc   = (const float*)d_in[2];
    const int*   ei    = (const int*)d_in[3];
    const unsigned char* amask = (const unsigned char*)d_in[4];
    const float* W_ep = (const float*)d_in[5],  *b_ep = (const float*)d_in[6];
    const float* W_pp = (const float*)d_in[7],  *b_pp = (const float*)d_in[8];
    const float* W_dp = (const float*)d_in[9],  *b_dp = (const float*)d_in[10];
    const float* W1  = (const float*)d_in[11], *as1 = (const float*)d_in[12];
    const float* ad1 = (const float*)d_in[13], *b1  = (const float*)d_in[14];
    const float* W2  = (const float*)d_in[15], *as2 = (const float*)d_in[16];
    const float* ad2 = (const float*)d_in[17], *b2  = (const float*)d_in[18];
    const float* W3  = (const float*)d_in[19], *as3 = (const float*)d_in[20];
    const float* ad3 = (const float*)d_in[21], *b3  = (const float*)d_in[22];
    const float* W_l1a = (const float*)d_in[23], *b_l1a = (const float*)d_in[24];
    const float* W_l1b = (const float*)d_in[25], *b_l1b = (const float*)d_in[26];
    const float* W_c1  = (const float*)d_in[27], *b_c1  = (const float*)d_in[28];
    const float* W_c2  = (const float*)d_in[29], *b_c2  = (const float*)d_in[30];

    float* ws = (float*)d_ws;
    float*    X    = ws;                    // NN*8
    float*    XN   = X    + (size_t)NN * 8; // NN*8
    float*    XH   = XN   + (size_t)NN * 8; // NN*8
    float*    ALS  = XH   + (size_t)NN * 8; // NN*4
    float*    ALD  = ALS  + (size_t)NN * 4; // NN*4
    unsigned* M    = (unsigned*)(ALD + (size_t)NN * 4); // NN*4
    float*    S    = (float*)M + (size_t)NN * 4;        // NN*4
    float*    ACC  = S    + (size_t)NN * 4; // NN*8
    float*    X3   = ACC  + (size_t)NN * 8; // NN
    float*    T1   = X3   + NN;             // 4096
    float*    C1   = T1   + 4096;           // 2048
    float*    HH   = C1   + 2048;           // 2048
    float*    PART = HH   + 2048;           // 20*4096

    float* out = (float*)d_out;             // [2048 probs][1 value][1 action]

    const int NB = (NN + 255) / 256;
    const int EB = (ETOT + 255) / 256;

    // ---- embedding ----
    k_embed<<<NB, 256, 0, stream>>>(proc, pat, doc, W_ep, b_ep, W_pp, b_pp, W_dp, b_dp, X);

    // ---- GAT layer 1: in=4, H=4, C=2 ----
    k_node_prep<<<NB, 256, 0, stream>>>(X, W1, as1, ad1, XH, ALS, ALD, M, S, ACC, 4, 4, 2);
    k_edge_max<<<EB, 256, 0, stream>>>(ei, ALS, ALD, M, 4);
    k_edge_acc<<<EB, 256, 0, stream>>>(ei, ALS, ALD, M, XH, S, ACC, 4, 2);
    k_finalize<<<NB, 256, 0, stream>>>(ACC, S, b1, XN, 4, 2, 1, 0x11110000u, 0);

    // ---- GAT layer 2: in=8, H=2, C=4 ----
    k_node_prep<<<NB, 256, 0, stream>>>(XN, W2, as2, ad2, XH, ALS, ALD, M, S, ACC, 8, 2, 4);
    k_edge_max<<<EB, 256, 0, stream>>>(ei, ALS, ALD, M, 2);
    k_edge_acc<<<EB, 256, 0, stream>>>(ei, ALS, ALD, M, XH, S, ACC, 2, 4);
    k_finalize<<<NB, 256, 0, stream>>>(ACC, S, b2, X, 2, 4, 1, 0x22220000u, 0);

    // ---- GAT layer 3: in=8, H=1, C=1 -> dense x3[N] ----
    k_node_prep<<<NB, 256, 0, stream>>>(X, W3, as3, ad3, XH, ALS, ALD, M, S, ACC, 8, 1, 1);
    k_edge_max<<<EB, 256, 0, stream>>>(ei, ALS, ALD, M, 1);
    k_edge_acc<<<EB, 256, 0, stream>>>(ei, ALS, ALD, M, XH, S, ACC, 1, 1);
    k_finalize<<<NB, 256, 0, stream>>>(ACC, S, b3, X3, 1, 1, 0, 0u, 1);

    // ---- policy head: t1 = x3 @ W_l1a + b  (20000 -> 4096), 20 slices x 1000 rows
    k_gemv_wmma<<<dim3(4096 / 64, 20), 128, 0, stream>>>(X3, W_l1a, PART, 4096, 1000);
    k_reduce<<<(4096 + 255) / 256, 256, 0, stream>>>(PART, b_l1a, T1, 4096, 20, 0);

    // ---- h = tanh(t1 @ W_l1b + b)  (4096 -> 2048), 8 slices x 512 rows
    k_gemv_wmma<<<dim3(2048 / 64, 8), 128, 0, stream>>>(T1, W_l1b, PART, 2048, 512);
    k_reduce<<<(2048 + 255) / 256, 256, 0, stream>>>(PART, b_l1b, HH, 2048, 8, 1);

    // ---- critic: c1 = x3 @ W_c1 + b  (20000 -> 2048), 20 slices x 1000 rows
    k_gemv_wmma<<<dim3(2048 / 64, 20), 128, 0, stream>>>(X3, W_c1, PART, 2048, 1000);
    k_reduce<<<(2048 + 255) / 256, 256, 0, stream>>>(PART, b_c1, C1, 2048, 20, 0);

    // ---- value = c1 . W_c2 + b_c2 ----
    k_value<<<1, 256, 0, stream>>>(C1, W_c2, b_c2, out + RR);

    // ---- probs + sampled action ----
    k_policy<<<1, 256, 0, stream>>>(HH, amask, out, out + RR + 1);
}